// GraphConvolutionalNetwork_59725815218350
// MI455X (gfx1250) — compile-verified
//
#include <hip/hip_runtime.h>
#include <hip/hip_bf16.h>

typedef float v2f __attribute__((ext_vector_type(2)));
typedef float v8f __attribute__((ext_vector_type(8)));

#define DFEAT 128
#define ALDS_STRIDE 132   // 128 + 4 pad: rows land on distinct banks

// ---------------------------------------------------------------------------
// C[M x N] = A[M x 128] @ W[128 x N], optional fused bias + relu.
// Block: 256 threads = 8 waves. Each block: FIVE 16-row tiles (80 rows,
// 50000/80 = 625 blocks exactly) x 8 col-tiles of 16 (128 cols);
// blockIdx.y selects the 128-col slab for N=256.
// Each wave reuses its B fragment across 5 WMMAs (A frags from LDS).
// N is a compile-time constant so B addressing folds into load immediates.
// ---------------------------------------------------------------------------
template <int N>
__global__ void gemm_k128_wmma(const float* __restrict__ A,
                               const float* __restrict__ W,
                               float* __restrict__ C,
                               const float* __restrict__ bias,
                               int do_relu)
{
    constexpr int MT = 5;                       // 16-row tiles per block
    __shared__ float Alds[MT * 16 * ALDS_STRIDE];
    const int tid = threadIdx.x;
    const int m0  = blockIdx.x * (MT * 16);

    // Cooperative load of the 80x128 A tile (10 float4 per thread, coalesced).
    #pragma unroll
    for (int it = 0; it < 10; ++it) {
        const int idx = it * 256 + tid;         // float4 index, 2560 total
        const int row = idx >> 5;               // 32 float4 per row
        const int c4  = (idx & 31) << 2;        // float column 0,4,...,124
        float4 v = *(const float4*)&A[(size_t)(m0 + row) * DFEAT + c4];
        *(float4*)&Alds[row * ALDS_STRIDE + c4] = v;  // 528B row pitch: 16B aligned
    }
    __syncthreads();

    const int wave = tid >> 5;
    const int lane = tid & 31;
    const int half = lane >> 4;   // K-half selector (ISA fragment layout)
    const int rm   = lane & 15;   // row (A) / col (B) within 16
    const int n0   = blockIdx.y * 128 + wave * 16;

    v8f acc[MT] = {};
    // B fragment base: lanes 0-15 read rows {kk,kk+1}, lanes 16-31 {kk+2,kk+3}
    const float* wbase = &W[(size_t)(2 * half) * N + n0 + rm];

    #pragma unroll
    for (int kk = 0; kk < 128; kk += 4) {
        v2f b;
        b.x = wbase[(size_t)(kk + 0) * N];
        b.y = wbase[(size_t)(kk + 1) * N];
        #pragma unroll
        for (int t = 0; t < MT; ++t) {
            const float* arow = &Alds[(t * 16 + rm) * ALDS_STRIDE + kk + 2 * half];
            v2f a;
            a.x = arow[0];
            a.y = arow[1];
            acc[t] = __builtin_amdgcn_wmma_f32_16x16x4_f32(
                         false, a, false, b, (short)0, acc[t], false, false);
        }
    }

    // C/D layout: VGPR r, lanes 0-15 -> M = r, lanes 16-31 -> M = r + 8
    const int col = n0 + rm;
    float bv = bias ? bias[col] : 0.0f;
    #pragma unroll
    for (int t = 0; t < MT; ++t) {
        #pragma unroll
        for (int r = 0; r < 8; ++r) {
            const int row = m0 + t * 16 + r + 8 * half;
            float v = acc[t][r] + bv;
            if (do_relu) v = v > 0.0f ? v : 0.0f;
            C[(size_t)row * N + col] = v;
        }
    }
}

// --- degree / norm ----------------------------------------------------------
__global__ void init_deg(float* __restrict__ deg, int n) {
    int i = blockIdx.x * blockDim.x + threadIdx.x;
    if (i < n) deg[i] = 1.0f;                 // self-loop contribution
}

__global__ void edge_deg(float* __restrict__ deg, const int* __restrict__ dst, int E) {
    int e = blockIdx.x * blockDim.x + threadIdx.x;
    if (e < E) unsafeAtomicAdd(&deg[dst[e]], 1.0f);
}

__global__ void rsqrt_inplace(float* __restrict__ d, int n) {
    int i = blockIdx.x * blockDim.x + threadIdx.x;
    if (i < n) d[i] = rsqrtf(d[i]);           // deg >= 1 always (self-loops)
}

// --- aggregation ------------------------------------------------------------
// H[i][:] = T[i][:] * dis[i]^2   (self-loop term; also initializes H)
__global__ void self_scale(float* __restrict__ H, const float* __restrict__ T,
                           const float* __restrict__ dis, int total4) {
    int i = blockIdx.x * blockDim.x + threadIdx.x;
    if (i >= total4) return;
    const int node = i >> 5;                  // 32 float4 chunks per node
    const float d = dis[node];
    const float s = d * d;
    float4 t = ((const float4*)T)[i];
    float4 r = make_float4(t.x * s, t.y * s, t.z * s, t.w * s);
    ((float4*)H)[i] = r;
}

// H[dst] += dis[src]*dis[dst] * T[src] ; one wave per edge, float4 per lane
__global__ void edge_agg(float* __restrict__ H, const float* __restrict__ T,
                         const float* __restrict__ dis,
                         const int* __restrict__ src, const int* __restrict__ dst,
                         int E) {
    const int e    = (blockIdx.x * blockDim.x + threadIdx.x) >> 5;
    const int lane = threadIdx.x & 31;
    if (e >= E) return;
    const int s = src[e];
    const int d = dst[e];
    const float nrm = dis[s] * dis[d];
    const float4 tv = *(const float4*)&T[(size_t)s * DFEAT + lane * 4];
    float* base = &H[(size_t)d * DFEAT + lane * 4];
    unsafeAtomicAdd(base + 0, tv.x * nrm);
    unsafeAtomicAdd(base + 1, tv.y * nrm);
    unsafeAtomicAdd(base + 2, tv.z * nrm);
    unsafeAtomicAdd(base + 3, tv.w * nrm);
}

// H[i][f] += b[f]; optional relu (feature width 128)
__global__ void bias_act(float* __restrict__ H, const float* __restrict__ b,
                         int total, int do_relu) {
    int i = blockIdx.x * blockDim.x + threadIdx.x;
    if (i >= total) return;
    float v = H[i] + b[i & (DFEAT - 1)];
    if (do_relu) v = v > 0.0f ? v : 0.0f;
    H[i] = v;
}

// out[n][o] = h3[n][:] . Wm2[:][o] + bm2[o]   (256 -> 10)
__global__ void head10(const float* __restrict__ H3, const float* __restrict__ Wm2,
                       const float* __restrict__ bm2, float* __restrict__ out,
                       int n_nodes) {
    int idx = blockIdx.x * blockDim.x + threadIdx.x;
    if (idx >= n_nodes * 10) return;
    const int node = idx / 10;
    const int o    = idx - node * 10;
    const float* h = &H3[(size_t)node * 256];
    float s = bm2[o];
    #pragma unroll 8
    for (int k = 0; k < 256; ++k)
        s = fmaf(h[k], Wm2[k * 10 + o], s);
    out[idx] = s;
}

// ---------------------------------------------------------------------------
extern "C" void kernel_launch(void* const* d_in, const int* in_sizes, int n_in,
                              void* d_out, int out_size, void* d_ws, size_t ws_size,
                              hipStream_t stream) {
    (void)n_in; (void)out_size; (void)ws_size;

    const float* x   = (const float*)d_in[0];
    const int*   ei  = (const int*)  d_in[1];
    const float* W0  = (const float*)d_in[2];
    const float* b0  = (const float*)d_in[3];
    const float* W1  = (const float*)d_in[4];
    const float* b1  = (const float*)d_in[5];
    const float* Wm1 = (const float*)d_in[6];
    const float* bm1 = (const float*)d_in[7];
    const float* Wm2 = (const float*)d_in[8];
    const float* bm2 = (const float*)d_in[9];

    const int nN = in_sizes[0] / DFEAT;   // 50000 (multiple of 80)
    const int E  = in_sizes[1] / 2;       // 800000
    const int* src = ei;
    const int* dst = ei + E;

    // Workspace layout (all 256B aligned):
    //   dis  : nN floats
    //   bufA : nN*128 floats  (t0 / t1; later first half of h3)
    //   bufB : nN*128 floats  (h;       later second half of h3 - contiguous)
    //   bufC : nN*128 floats  (h2)
    char* ws = (char*)d_ws;
    auto al = [](size_t v) { return (v + 255) & ~(size_t)255; };
    size_t off = 0;
    float* dis  = (float*)(ws + off); off = al(off + (size_t)nN * 4);
    float* bufA = (float*)(ws + off); off = al(off + (size_t)nN * DFEAT * 4);
    float* bufB = (float*)(ws + off); off = al(off + (size_t)nN * DFEAT * 4);
    float* bufC = (float*)(ws + off); off = al(off + (size_t)nN * DFEAT * 4);
    float* h3   = bufA;  // needs nN*256 floats = bufA+bufB (contiguous, both dead)

    const int T = 256;
    const int totalF  = nN * DFEAT;        // elementwise over features
    const int total4  = totalF / 4;        // float4 chunks
    const int mBlocks = nN / 80;           // 625 (5 tiles of 16 rows per block)

    // normalization coefficients
    init_deg<<<(nN + T - 1) / T, T, 0, stream>>>(dis, nN);
    edge_deg<<<(E + T - 1) / T, T, 0, stream>>>(dis, dst, E);
    rsqrt_inplace<<<(nN + T - 1) / T, T, 0, stream>>>(dis, nN);

    // layer 0: t0 = x @ W0 ; h = scatter(norm * t0) + b0 ; relu
    gemm_k128_wmma<128><<<dim3(mBlocks, 1), T, 0, stream>>>(x, W0, bufA, nullptr, 0);
    self_scale<<<(total4 + T - 1) / T, T, 0, stream>>>(bufB, bufA, dis, total4);
    edge_agg<<<((size_t)E * 32 + T - 1) / T, T, 0, stream>>>(bufB, bufA, dis, src, dst, E);
    bias_act<<<(totalF + T - 1) / T, T, 0, stream>>>(bufB, b0, totalF, 1);

    // layer 1: t1 = h @ W1 ; h2 = scatter(norm * t1) + b1
    gemm_k128_wmma<128><<<dim3(mBlocks, 1), T, 0, stream>>>(bufB, W1, bufA, nullptr, 0);
    self_scale<<<(total4 + T - 1) / T, T, 0, stream>>>(bufC, bufA, dis, total4);
    edge_agg<<<((size_t)E * 32 + T - 1) / T, T, 0, stream>>>(bufC, bufA, dis, src, dst, E);
    bias_act<<<(totalF + T - 1) / T, T, 0, stream>>>(bufC, b1, totalF, 0);

    // MLP: h3 = relu(h2 @ Wm1 + bm1)   (128 -> 256, bias+relu fused)
    gemm_k128_wmma<256><<<dim3(mBlocks, 2), T, 0, stream>>>(bufC, Wm1, h3, bm1, 1);

    // head: out = h3 @ Wm2 + bm2       (256 -> 10)
    head10<<<(nN * 10 + T - 1) / T, T, 0, stream>>>(h3, Wm2, bm2, (float*)d_out, nN);
}